// Histogram2D_wrap_48558900249125
// MI455X (gfx1250) — compile-verified
//
#include <hip/hip_runtime.h>
#include <math.h>

// ---- constants matching the reference ----
#define BINS   100
#define NPB    (BINS * BINS)          // 10000 bins per batch
#define DELTA  0.01f                  // (MAX-MIN)/BINS
#define EPS    1e-5f
#define TILE   1024                   // points staged in LDS per step

typedef float v2f __attribute__((ext_vector_type(2)));
typedef float v8f __attribute__((ext_vector_type(8)));

// --------------------------------------------------------------------------
// Kernel 1: raw soft-histogram counts (gather formulation, deterministic).
//
// Thread mapping: blockDim = 256 = 2 rows x 128 lanes.
//   bcol = tid & 127  (v-bin, lanes 0..99 active)
//   a    = rowbase + (tid >> 7)  (u-bin) -> every wave sits in ONE row, so
//   du = |u - ca| and the (du < 2*DELTA) early-out are wave-uniform: ~96% of
//   points are rejected with a scalar branch, no per-lane work.
// Points are staged in LDS (broadcast reads, no bank conflicts).
// --------------------------------------------------------------------------
__global__ __launch_bounds__(256)
void hist2d_counts_kernel(const float* __restrict__ x,
                          float* __restrict__ out, int N)
{
    __shared__ float2 tile[TILE];

    const int batch   = blockIdx.x / (BINS / 2);        // 2 rows per block
    const int rowbase = (blockIdx.x % (BINS / 2)) * 2;
    const int tid     = threadIdx.x;
    const int r       = tid >> 7;                        // 0..1
    const int bcol    = tid & 127;                       // 0..127
    const int a       = rowbase + r;
    const bool active = (bcol < BINS);

    const float ca = (a + 0.5f) * DELTA;                 // same f32 expr as ref
    const float cb = (bcol + 0.5f) * DELTA;

    const float2* xb = (const float2*)(x) + (size_t)batch * N;

    float acc = 0.0f;

    for (int t0 = 0; t0 < N; t0 += TILE) {
        const int tpts = (N - t0 < TILE) ? (N - t0) : TILE;
        __syncthreads();                                 // prev tile consumed
        // stage + wrap into [0,1): x - floor(x) == jnp.remainder(x, 1.0)
        // (+-inf wraps to NaN, exactly like the reference's wrapped values)
        for (int k = tid; k < TILE; k += 256) {
            const int idx = t0 + k;
            float2 p;
            if (idx < N) {
                p = xb[idx];
                p.x = p.x - floorf(p.x);
                p.y = p.y - floorf(p.y);
            } else {
                p.x = __builtin_nanf("");
                p.y = __builtin_nanf("");
            }
            tile[k] = p;
        }
        __syncthreads();

        for (int p = 0; p < tpts; ++p) {
            const float u = tile[p].x;                   // LDS broadcast
            const float v = tile[p].y;
            // post-wrap finiteness mask (NaN here <=> ref's ~isfinite)
            if (!(u == u) || !(v == v)) continue;        // uniform branch
            const float du = fabsf(u - ca);              // wave-uniform
            if (du < 2.0f * DELTA) {                     // uniform early-out
                const float dv = fabsf(v - cb);
                const float w  = DELTA - 0.5f * (du + dv);
                acc += fmaxf(w, 0.0f);                   // relu, same as ref
            }
            // if du >= 2*DELTA then du+dv >= 2*DELTA -> w <= 0 exactly in
            // f32 (rounding is monotone, *0.5 exact), so skipping is lossless
        }
    }

    if (active)
        out[(size_t)batch * NPB + a * BINS + bcol] = acc;
}

// --------------------------------------------------------------------------
// Kernel 2: normalize each batch histogram: out /= (sum(out) + EPS).
// The 10000-element reduction is done with chained V_WMMA_F32_16X16X4_F32:
// B = ones(4x16), so D accumulates row-sums of A; sum(all 256 D entries)
// = 16 * sum(all loaded values), regardless of how values map into A's
// VGPR layout. That freedom lets lane l load a CONTIGUOUS float2
// (global_load_b64, fully coalesced, no bounds check) for the 156 full
// 64-value chunks; only the final 16-value tail is guarded.
// One wave (32 lanes, EXEC all-1s) per batch. Deterministic.
// --------------------------------------------------------------------------
__global__ __launch_bounds__(32)
void hist2d_normalize_kernel(float* __restrict__ out)
{
    const int   batch = blockIdx.x;
    float*      h     = out + (size_t)batch * NPB;
    const int   lane  = threadIdx.x;                     // 0..31, all active

    v8f acc = {0.f, 0.f, 0.f, 0.f, 0.f, 0.f, 0.f, 0.f};
    v2f ones;
    ones.x = 1.0f;
    ones.y = 1.0f;

    const int FULL = NPB / 64;                           // 156 full chunks
    const float2* h2 = (const float2*)h;
    for (int c = 0; c < FULL; ++c) {
        const float2 p = h2[c * 32 + lane];              // contiguous b64 load
        v2f av;
        av.x = p.x;
        av.y = p.y;
        // D = A x ones + C  (8 args: neg_a, A, neg_b, B, c_mod, C, reuse_a, reuse_b)
        acc = __builtin_amdgcn_wmma_f32_16x16x4_f32(
                  false, av, false, ones, (short)0, acc, false, false);
    }
    {   // tail: NPB - FULL*64 = 16 values (even count, so pairs stay aligned)
        const int i0 = FULL * 64 + 2 * lane;
        v2f av;
        av.x = 0.0f;
        av.y = 0.0f;
        if (i0 < NPB) {
            const float2 p = h2[i0 >> 1];
            av.x = p.x;
            av.y = p.y;
        }
        acc = __builtin_amdgcn_wmma_f32_16x16x4_f32(
                  false, av, false, ones, (short)0, acc, false, false);
    }

    float partial = acc[0] + acc[1] + acc[2] + acc[3]
                  + acc[4] + acc[5] + acc[6] + acc[7];
#pragma unroll
    for (int off = 16; off > 0; off >>= 1)
        partial += __shfl_xor(partial, off, 32);

    // every A value was counted 16x (16 columns of ones)
    const float s = partial * (1.0f / 16.0f) + EPS;

    for (int i = lane; i < NPB; i += 32)
        h[i] = h[i] / s;                                 // IEEE divide, like ref
}

// --------------------------------------------------------------------------
extern "C" void kernel_launch(void* const* d_in, const int* in_sizes, int n_in,
                              void* d_out, int out_size, void* d_ws, size_t ws_size,
                              hipStream_t stream)
{
    (void)n_in; (void)out_size; (void)d_ws; (void)ws_size;
    const float* x   = (const float*)d_in[0];   // (2, N, 2) f32
    float*       out = (float*)d_out;           // (2, 100, 100) f32
    const int    N   = in_sizes[0] / 4;         // B=2, 2 coords per point

    // K1: 2 batches * 50 blocks (2 u-rows each) = 100 blocks
    hist2d_counts_kernel<<<dim3(100), dim3(256), 0, stream>>>(x, out, N);
    // K2: one wave per batch, WMMA reduction + in-place normalize
    hist2d_normalize_kernel<<<dim3(2), dim3(32), 0, stream>>>(out);
}